// BoxTransformerEncoder_420906795020
// MI455X (gfx1250) — compile-verified
//
#include <hip/hip_runtime.h>
#include <hip/hip_bf16.h>
#include <math.h>

// Problem constants from the reference.
#define BB 8
#define NN 20000
#define DD 256
#define KK 300
#define NEGV (-65504.0f)

typedef __attribute__((ext_vector_type(2))) float v2f;
typedef __attribute__((ext_vector_type(4))) float v4f;
typedef __attribute__((ext_vector_type(8))) float v8f;

// ---------------------------------------------------------------------------
// Kernel 1: fused src copy + logits (+mask) -> sortable u32 keys.
// One wave (32 lanes) per row of 256 floats. The copy output is written with
// non-temporal stores (TH=NT): it is never re-read on device, so keep the
// 164MB stream out of L2 and preserve residency for weights/keys.
// ---------------------------------------------------------------------------
__device__ __forceinline__ unsigned sortable_key(float f) {
  unsigned u = __float_as_uint(f);
  return (u & 0x80000000u) ? ~u : (u | 0x80000000u);
}

__global__ void __launch_bounds__(256)
k1_copy_logits(const float* __restrict__ src,
               const float* __restrict__ refw,
               const unsigned char* __restrict__ smask,
               const float* __restrict__ cw,
               const float* __restrict__ cb,
               float* __restrict__ out0,
               unsigned* __restrict__ keys) {
  int row = blockIdx.x * 8 + (threadIdx.x >> 5);
  int lane = threadIdx.x & 31;
  if (row >= BB * NN) return;

  const v4f* s4 = (const v4f*)(src + (size_t)row * DD);
  v4f* o4 = (v4f*)(out0 + (size_t)row * DD);
  const v4f* w4 = (const v4f*)cw;

  float partial = 0.f;
#pragma unroll
  for (int q = 0; q < 2; ++q) {
    int v = lane * 2 + q;           // v4f index 0..63
    v4f x = s4[v];
    __builtin_nontemporal_store(x, &o4[v]);  // identity "encoder output"
    v4f w = w4[v];
    partial += x.x * w.x + x.y * w.y + x.z * w.z + x.w * w.w;
  }
#pragma unroll
  for (int off = 16; off >= 1; off >>= 1)
    partial += __shfl_xor(partial, off, 32);

  if (lane == 0) {
    float logit = partial + cb[0];
    float x = refw[(size_t)row * 4 + 0];
    float y = refw[(size_t)row * 4 + 1];
    bool valid = (x > 0.01f) && (x < 0.99f) && (y > 0.01f) && (y < 0.99f);
    bool m = (smask[row] != 0) && !valid;
    if (m) logit = NEGV;
    keys[row] = sortable_key(logit);
  }
}

// ---------------------------------------------------------------------------
// Kernel 2: per-batch exact top-K via 32-bit radix-select + exact ranking.
// One 1024-thread block per batch. Reproduces lax.top_k order:
// descending value, ties broken by ascending index.
// ---------------------------------------------------------------------------
#define LIST_CAP 2048

__global__ void __launch_bounds__(1024)
k2_topk(const unsigned* __restrict__ keys, int* __restrict__ idx) {
  int b = blockIdx.x;
  const unsigned* kb = keys + (size_t)b * NN;
  __shared__ unsigned s_cnt;
  __shared__ int s_list_n;
  __shared__ int s_list[LIST_CAP];
  int tid = threadIdx.x;

  // Find T = K-th largest key by bitwise prefix refinement.
  unsigned prefix = 0u;
  for (int bit = 31; bit >= 0; --bit) {
    unsigned trial = prefix | (1u << bit);
    if (tid == 0) s_cnt = 0u;
    __syncthreads();
    unsigned local = 0u;
    for (int n = tid; n < NN; n += 1024) local += (kb[n] >= trial) ? 1u : 0u;
    if (local) atomicAdd(&s_cnt, local);
    __syncthreads();
    if (s_cnt >= (unsigned)KK) prefix = trial;
    __syncthreads();
  }
  unsigned T = prefix;

  // Collect candidates (key >= T), count strictly-greater G.
  if (tid == 0) { s_cnt = 0u; s_list_n = 0; }
  __syncthreads();
  unsigned localg = 0u;
  for (int n = tid; n < NN; n += 1024) {
    unsigned kk = kb[n];
    if (kk > T) localg++;
    if (kk >= T) {
      int p = atomicAdd(&s_list_n, 1);
      if (p < LIST_CAP) s_list[p] = n;
    }
  }
  if (localg) atomicAdd(&s_cnt, localg);
  __syncthreads();
  int G = (int)s_cnt;
  int M = s_list_n < LIST_CAP ? s_list_n : LIST_CAP;

  // Exact rank for each candidate -> scatter index into sorted slot.
  for (int c = tid; c < M; c += 1024) {
    int i = s_list[c];
    unsigned ki = kb[i];
    int rank;
    if (ki > T) {
      int gt = 0, eqb = 0;
      for (int j = 0; j < NN; ++j) {
        unsigned kj = kb[j];
        gt += (kj > ki) ? 1 : 0;
        eqb += ((kj == ki) && (j < i)) ? 1 : 0;
      }
      rank = gt + eqb;
    } else {  // ki == T
      int eqb = 0;
      for (int j = 0; j < i; ++j) eqb += (kb[j] == T) ? 1 : 0;
      rank = G + eqb;
    }
    if (rank < KK) idx[b * KK + rank] = i;
  }
}

// ---------------------------------------------------------------------------
// Kernel 3: gather 16 selected rows, run 256x256 GEMMs with f32 WMMA,
// LayerNorm, bbox head + sigmoid, positional embedding.
// Block = 512 threads = 16 waves; one wave per 16-column output tile.
// ---------------------------------------------------------------------------
#define XS 260  // LDS row stride in floats (260 % 64 == 4 -> conflict-free)

__device__ __forceinline__ v8f wmma_tile(const float* __restrict__ sX,
                                         const float* __restrict__ W,
                                         const float* __restrict__ bias,
                                         int nb, int lane) {
  int half = lane >> 4;
  int m = lane & 15;
  float bv = bias[nb * 16 + m];  // C/D layout: all 8 regs of a lane share N=m
  v8f acc;
#pragma unroll
  for (int r = 0; r < 8; ++r) acc[r] = bv;

  const float* xrow = sX + m * XS + 2 * half;            // A[m][4k+2h+{0,1}]
  const float* wcol = W + (2 * half) * DD + nb * 16 + m; // B[4k+2h+{0,1}][col]
  for (int kb2 = 0; kb2 < 16; ++kb2) {
    // Speculative prefetch of the weight stream two k-groups (32KB) ahead;
    // past-the-end prefetches are silently dropped (ISA 10.5).
    __builtin_prefetch(wcol + (size_t)(16 * (kb2 + 2)) * DD, 0, 1);
#pragma unroll
    for (int ki = 0; ki < 4; ++ki) {
      int k = kb2 * 4 + ki;
      v2f a, bfr;
      a.x = xrow[4 * k];
      a.y = xrow[4 * k + 1];
      const float* wp = wcol + (size_t)(4 * k) * DD;
      bfr.x = wp[0];
      bfr.y = wp[DD];
      acc = __builtin_amdgcn_wmma_f32_16x16x4_f32(false, a, false, bfr,
                                                  (short)0, acc, false, false);
    }
  }
  return acc;
}

__device__ __forceinline__ void store_tile(float* __restrict__ sOut, v8f acc,
                                           int nb, int lane, bool relu) {
  int half = lane >> 4;
  int m = lane & 15;
#pragma unroll
  for (int r = 0; r < 8; ++r) {
    float v = acc[r];
    if (relu) v = fmaxf(v, 0.f);
    sOut[(r + 8 * half) * XS + nb * 16 + m] = v;
  }
}

__global__ void __launch_bounds__(512)
k3_head(const float* __restrict__ src, const float* __restrict__ refw,
        const int* __restrict__ g_idx,
        const float* __restrict__ enc_w, const float* __restrict__ enc_b,
        const float* __restrict__ ln_g, const float* __restrict__ ln_b,
        const float* __restrict__ bb_w1, const float* __restrict__ bb_b1,
        const float* __restrict__ bb_w2, const float* __restrict__ bb_b2,
        const float* __restrict__ bb_w3, const float* __restrict__ bb_b3,
        float* __restrict__ out_embed, float* __restrict__ out_rw,
        float* __restrict__ out_pos) {
  __shared__ float sA[16 * XS];   // gathered X
  __shared__ float sB[16 * XS];   // enc output -> later H2
  __shared__ float sC[16 * XS];   // H1
  __shared__ float s_rw[16 * 4];
  __shared__ float s_orw[16 * 4];

  int t = blockIdx.x;             // 0..149, 16 rows each
  int tid = threadIdx.x;
  int wave = tid >> 5;
  int lane = tid & 31;

  // ---- gather X rows + ref windows ----
  {
    int i = wave;                 // wave per row
    int flat = t * 16 + i;
    int b = flat / KK, k = flat % KK;
    int n = g_idx[b * KK + k];
    const v4f* s4 = (const v4f*)(src + ((size_t)b * NN + n) * DD);
    v4f* d4 = (v4f*)(sA + i * XS);
    d4[lane * 2] = s4[lane * 2];
    d4[lane * 2 + 1] = s4[lane * 2 + 1];
    if (lane < 4) s_rw[i * 4 + lane] = refw[((size_t)b * NN + n) * 4 + lane];
  }
  __syncthreads();

  // ---- GEMM1 (enc) and GEMM2 (bbox layer 1, relu) ----
  {
    int nb = wave;
    v8f e = wmma_tile(sA, enc_w, enc_b, nb, lane);
    store_tile(sB, e, nb, lane, false);
    v8f h = wmma_tile(sA, bb_w1, bb_b1, nb, lane);
    store_tile(sC, h, nb, lane, true);
  }
  __syncthreads();

  // ---- LayerNorm of sB -> out_embed; wave i handles row i ----
  {
    int i = wave;
    const float* y = sB + i * XS;
    float s = 0.f, sq = 0.f;
#pragma unroll
    for (int q = 0; q < 8; ++q) {
      float v = y[lane * 8 + q];
      s += v;
      sq += v * v;
    }
#pragma unroll
    for (int off = 16; off >= 1; off >>= 1) {
      s += __shfl_xor(s, off, 32);
      sq += __shfl_xor(sq, off, 32);
    }
    float mean = s * (1.f / 256.f);
    float var = sq * (1.f / 256.f) - mean * mean;
    float inv = rsqrtf(var + 1e-5f);
    float* o = out_embed + (size_t)(t * 16 + i) * DD;
#pragma unroll
    for (int q = 0; q < 8; ++q) {
      int c = lane * 8 + q;
      o[c] = (y[c] - mean) * inv * ln_g[c] + ln_b[c];
    }
  }
  __syncthreads();

  // ---- GEMM3: H2 = relu(H1 @ bb_w2 + b2); sC -> sB ----
  {
    int nb = wave;
    v8f h2 = wmma_tile(sC, bb_w2, bb_b2, nb, lane);
    store_tile(sB, h2, nb, lane, true);
  }
  __syncthreads();

  // ---- bbox: tmp = H2 @ bb_w3 + b3 + inv_sigmoid(rw); sigmoid ----
  if (tid < 64) {
    int row = tid >> 2, col = tid & 3;
    float s = bb_b3[col];
    const float* h = sB + row * XS;
    for (int d0 = 0; d0 < DD; ++d0) s += h[d0] * bb_w3[d0 * 4 + col];
    float x = s_rw[row * 4 + col];
    x = fminf(fmaxf(x, 0.f), 1.f);
    float x1 = fmaxf(x, 1e-5f);
    float x2 = fmaxf(1.f - x, 1e-5f);
    s += logf(x1 / x2);
    float orw = 1.f / (1.f + expf(-s));
    out_rw[(size_t)(t * 16 + row) * 4 + col] = orw;
    s_orw[row * 4 + col] = orw;
  }
  __syncthreads();

  // ---- positional embedding: pos(xy) + pos(wh) ----
  {
    const float TWO_PI = 6.283185307179586f;
    const float LOG1E4_OVER_64 = 9.210340371976184f / 64.f;
    for (int e = tid; e < 16 * 256; e += 512) {
      int row = e >> 8;
      int d0 = e & 255;
      int c = d0 >> 7;            // 0: x/w, 1: y/h
      int m = (d0 & 127) >> 1;    // frequency index
      int iscos = d0 & 1;
      float invdim = expf(-(float)m * LOG1E4_OVER_64);  // 10000^(-m/64)
      float v1 = s_orw[row * 4 + c] * TWO_PI * invdim;
      float v2 = s_orw[row * 4 + c + 2] * TWO_PI * invdim;
      float r = iscos ? (cosf(v1) + cosf(v2)) : (sinf(v1) + sinf(v2));
      out_pos[(size_t)(t * 16 + row) * DD + d0] = r;
    }
  }
}

// ---------------------------------------------------------------------------
extern "C" void kernel_launch(void* const* d_in, const int* in_sizes, int n_in,
                              void* d_out, int out_size, void* d_ws, size_t ws_size,
                              hipStream_t stream) {
  const float* src = (const float*)d_in[0];
  const float* refw = (const float*)d_in[1];
  const unsigned char* smask = (const unsigned char*)d_in[2];
  const float* cw = (const float*)d_in[3];
  const float* cb = (const float*)d_in[4];
  const float* enc_w = (const float*)d_in[5];
  const float* enc_b = (const float*)d_in[6];
  const float* ln_g = (const float*)d_in[7];
  const float* ln_b = (const float*)d_in[8];
  const float* bb_w1 = (const float*)d_in[9];
  const float* bb_b1 = (const float*)d_in[10];
  const float* bb_w2 = (const float*)d_in[11];
  const float* bb_b2 = (const float*)d_in[12];
  const float* bb_w3 = (const float*)d_in[13];
  const float* bb_b3 = (const float*)d_in[14];

  float* out0 = (float*)d_out;                              // [B,N,D]
  float* out_embed = out0 + (size_t)BB * NN * DD;           // [B,K,D]
  float* out_rw = out_embed + (size_t)BB * KK * DD;         // [B,K,4]
  float* out_pos = out_rw + (size_t)BB * KK * 4;            // [B,K,D]

  unsigned* keys = (unsigned*)d_ws;                         // B*N u32
  int* idx = (int*)((char*)d_ws + sizeof(unsigned) * (size_t)BB * NN);

  k1_copy_logits<<<(BB * NN) / 8, 256, 0, stream>>>(src, refw, smask, cw, cb,
                                                    out0, keys);
  k2_topk<<<BB, 1024, 0, stream>>>(keys, idx);
  k3_head<<<(BB * KK) / 16, 512, 0, stream>>>(
      src, refw, idx, enc_w, enc_b, ln_g, ln_b, bb_w1, bb_b1, bb_w2, bb_b2,
      bb_w3, bb_b3, out_embed, out_rw, out_pos);
}